// SO8Rotation_25838523252902
// MI455X (gfx1250) — compile-verified
//
#include <hip/hip_runtime.h>
#include <hip/hip_bf16.h>
#include <math.h>

#define ROT_DIM 8

typedef __attribute__((ext_vector_type(2))) float v2f;
typedef __attribute__((ext_vector_type(8))) float v8f;

// ---------------------------------------------------------------------------
// Prelude: compute the 8x8 rotation matrix R from rotation_params, exactly
// mirroring the reference: row-normalize twice, flip sign if det<0, scale by
// |det|^(-1/8). Single thread — 8x8 is trivial. Result -> d_ws (64 floats).
// ---------------------------------------------------------------------------
__global__ void so8_make_R(const float* __restrict__ P, float* __restrict__ R) {
    if (threadIdx.x != 0 || blockIdx.x != 0) return;

    float Q[ROT_DIM][ROT_DIM];
    // first row-normalize
    for (int i = 0; i < ROT_DIM; ++i) {
        float s = 0.f;
        for (int j = 0; j < ROT_DIM; ++j) { float v = P[i*ROT_DIM + j]; s += v*v; }
        float inv = 1.0f / sqrtf(s);
        for (int j = 0; j < ROT_DIM; ++j) Q[i][j] = P[i*ROT_DIM + j] * inv;
    }
    // second row-normalize (idempotent but faithful to reference)
    for (int i = 0; i < ROT_DIM; ++i) {
        float s = 0.f;
        for (int j = 0; j < ROT_DIM; ++j) s += Q[i][j]*Q[i][j];
        float inv = 1.0f / sqrtf(s);
        for (int j = 0; j < ROT_DIM; ++j) Q[i][j] *= inv;
    }
    // det via LU with partial pivoting (fp32)
    float M[ROT_DIM][ROT_DIM];
    for (int i = 0; i < ROT_DIM; ++i)
        for (int j = 0; j < ROT_DIM; ++j) M[i][j] = Q[i][j];
    float det = 1.0f;
    for (int k = 0; k < ROT_DIM; ++k) {
        int p = k;
        for (int r = k + 1; r < ROT_DIM; ++r)
            if (fabsf(M[r][k]) > fabsf(M[p][k])) p = r;
        if (p != k) {
            for (int c = 0; c < ROT_DIM; ++c) { float t = M[k][c]; M[k][c] = M[p][c]; M[p][c] = t; }
            det = -det;
        }
        det *= M[k][k];
        float inv = 1.0f / M[k][k];
        for (int r = k + 1; r < ROT_DIM; ++r) {
            float f = M[r][k] * inv;
            for (int c = k; c < ROT_DIM; ++c) M[r][c] -= f * M[k][c];
        }
    }
    float sgn = (det < 0.0f) ? -1.0f : 1.0f;
    det = fabsf(det);
    float scale = sgn * powf(det, -1.0f / (float)ROT_DIM);
    for (int i = 0; i < ROT_DIM; ++i)
        for (int j = 0; j < ROT_DIM; ++j)
            R[i*ROT_DIM + j] = Q[i][j] * scale;
}

// ---------------------------------------------------------------------------
// Main: streaming block-diagonal rotation via V_WMMA_F32_16X16X4_F32.
// Each wave owns 16x16 fp32 tiles (256 consecutive floats = two 8-blocks per
// row x 16 rows). out_tile(16x16) = x_tile(16x16) @ blockdiag(R,R), computed
// as 4 accumulating 16x16x4 fp32 WMMAs (K split 0..3, 4..7, 8..11, 12..15).
// Memory-bound: 268 MB total traffic -> ~11.5 us at 23.3 TB/s HBM.
// ---------------------------------------------------------------------------
__global__ void __launch_bounds__(256)
so8_rotate_wmma(const float* __restrict__ x, const float* __restrict__ R,
                float* __restrict__ out, int numTiles) {
    const int lane = threadIdx.x & 31;
    const int hi   = lane >> 4;      // 0: lanes 0-15, 1: lanes 16-31
    const int m    = lane & 15;      // row (A) / col (B,D) index
    const int wavesPerBlock = blockDim.x >> 5;
    const int wave   = blockIdx.x * wavesPerBlock + (threadIdx.x >> 5);
    const int nWaves = gridDim.x * wavesPerBlock;

    // Build B = blockdiag(R, R) in WMMA B-layout, once per wave.
    // B_k VGPR0 holds K rows {4k, 4k+2}, VGPR1 holds {4k+1, 4k+3} (cols across lanes).
    v2f b[4];
#pragma unroll
    for (int k = 0; k < 4; ++k) {
#pragma unroll
        for (int v = 0; v < 2; ++v) {
            int r = 4*k + v + 2*hi;              // K row within 16x16 blockdiag
            int c = m;                           // N column
            float val = ((r >> 3) == (c >> 3)) ? R[(r & 7) * ROT_DIM + (c & 7)] : 0.0f;
            b[k][v] = val;
        }
    }

    for (int t = wave; t < numTiles; t += nWaves) {
        const float* src = x + (size_t)t * 256;

        // A-layout: lane (m, hi) holds X[m][4k+2hi], X[m][4k+2hi+1] per k-step.
        v2f a[4];
#pragma unroll
        for (int k = 0; k < 4; ++k)
            a[k] = *(const v2f*)(src + m * 16 + 4 * k + 2 * hi);

        v8f acc = {};
#pragma unroll
        for (int k = 0; k < 4; ++k)
            acc = __builtin_amdgcn_wmma_f32_16x16x4_f32(
                /*neg_a=*/false, a[k], /*neg_b=*/false, b[k],
                /*c_mod=*/(short)0, acc, /*reuse_a=*/false, /*reuse_b=*/false);

        // D-layout: VGPR i -> rows {i, i+8}, columns across lanes. 8 coalesced b32 stores.
        float* dst = out + (size_t)t * 256;
#pragma unroll
        for (int i = 0; i < 8; ++i)
            dst[(i + 8 * hi) * 16 + m] = acc[i];
    }
}

extern "C" void kernel_launch(void* const* d_in, const int* in_sizes, int n_in,
                              void* d_out, int out_size, void* d_ws, size_t ws_size,
                              hipStream_t stream) {
    const float* x = (const float*)d_in[0];          // (4, 2048, 4096) fp32
    const float* P = (const float*)d_in[1];          // (8, 8) fp32 rotation_params
    float* out = (float*)d_out;
    float* Rws = (float*)d_ws;                       // 64 floats of scratch for R

    so8_make_R<<<1, 32, 0, stream>>>(P, Rws);

    // total elements = 33,554,432 -> 131,072 tiles of 256 floats (divisible).
    int numTiles = out_size / 256;
    int wavesPerBlock = 8;                           // 256 threads, wave32
    int tilesPerWave = 4;
    int blocks = (numTiles + wavesPerBlock * tilesPerWave - 1) / (wavesPerBlock * tilesPerWave);
    if (blocks < 1) blocks = 1;
    if (blocks > 16384) blocks = 16384;
    so8_rotate_wmma<<<blocks, 256, 0, stream>>>(x, Rws, out, numTiles);
}